// LinearAttention_5523327943104
// MI455X (gfx1250) — compile-verified
//
#include <hip/hip_runtime.h>

typedef __attribute__((ext_vector_type(2))) float v2f;
typedef __attribute__((ext_vector_type(8))) float v8f;
typedef __attribute__((address_space(3))) float lds_f;

// D = A(16x4, f32) * B(4x16, f32) + C(16x16, f32), full-precision matrix op.
#define WMMA_F32(a, b, c) \
  __builtin_amdgcn_wmma_f32_16x16x4_f32(false, (a), false, (b), (short)0, (c), false, false)

// Async global -> LDS copy (ASYNCcnt-tracked), GV addressing, 16B payload.
// VDST holds the LDS byte address (AS3 pointer = 32-bit LDS offset).
__device__ __forceinline__ void async_ld_b128(const float* g, float* l) {
  asm volatile("global_load_async_to_lds_b128 %0, %1, off"
               :
               : "v"((lds_f*)l), "v"(g)
               : "memory");
}

// Wait until at most N async loads remain outstanding. Memory clobber keeps
// the compiler from hoisting LDS reads above the wait (it cannot see the
// async writes to LDS).
template <int N>
__device__ __forceinline__ void wait_async() {
  asm volatile("s_wait_asynccnt %0" ::"i"(N) : "memory");
}

// Problem constants (from the reference)
static constexpr int NB    = 4;      // batch
static constexpr int NN    = 8192;   // sequence
static constexpr int DIMC  = 256;    // model dim
static constexpr int HEADS = 8;
static constexpr int DH    = 64;     // dim per head
static constexpr int INNER = 512;    // heads * dh
static constexpr int BH    = 32;     // batch * heads
static constexpr int BN    = 32768;  // batch * seq
static constexpr int NSPLIT = 16;    // split-K factor for k^T v

// ---------------------------------------------------------------------------
// Kernel 1: qkv = x @ Wqkv^T  -> scatter to Q/K/V in [bh][n][64] layout.
// Tile: 128 rows x 64 cols, 8 waves; double-buffered async global->LDS.
// Per chunk per thread: 8 (A) + 4 (B) = 12 async b128 loads.
// ---------------------------------------------------------------------------
__device__ __forceinline__ void qkv_issue(const float* __restrict__ x,
                                          const float* __restrict__ Wqkv,
                                          int row0, int f0, int kk,
                                          float (*As)[68], float (*Bs)[68],
                                          int tid) {
  for (int i = tid; i < 128 * 16; i += 256) {
    int r = i >> 4, c4 = (i & 15) << 2;
    async_ld_b128(&x[(size_t)(row0 + r) * DIMC + kk + c4], &As[r][c4]);
  }
  for (int i = tid; i < 64 * 16; i += 256) {
    int r = i >> 4, c4 = (i & 15) << 2;
    async_ld_b128(&Wqkv[(size_t)(f0 + r) * DIMC + kk + c4], &Bs[r][c4]);
  }
}

__global__ __launch_bounds__(256) void k_qkv(const float* __restrict__ x,
                                             const float* __restrict__ Wqkv,
                                             float* __restrict__ Qb,
                                             float* __restrict__ Kb,
                                             float* __restrict__ Vb) {
  __shared__ float As[2][128][68];
  __shared__ float Bs[2][64][68];
  const int tid  = threadIdx.x;
  const int lane = tid & 31;
  const int w    = tid >> 5;
  const int wr   = w >> 1;     // 0..3
  const int wc   = w & 1;      // 0..1
  const int ct   = blockIdx.x % 24;
  const int rt   = blockIdx.x / 24;
  const int row0 = rt * 128;
  const int f0   = ct * 64;

  const int khalf = (lane >> 4) << 1;  // 0 or 2 (K offset of this lane half)
  const int mrow  = lane & 15;

  v8f acc[2][2] = {};

  qkv_issue(x, Wqkv, row0, f0, 0, As[0], Bs[0], tid);

  for (int c = 0; c < 4; ++c) {
    const int cur = c & 1;
    if (c + 1 < 4) {
      // Prefetch next chunk into the other buffer (overlaps with WMMA below),
      // then drain down to the 12 just-issued loads -> current chunk landed.
      qkv_issue(x, Wqkv, row0, f0, (c + 1) * 64, As[cur ^ 1], Bs[cur ^ 1], tid);
      wait_async<12>();
    } else {
      wait_async<0>();
    }
    __syncthreads();

    for (int k4 = 0; k4 < 64; k4 += 4) {
      const int kq = k4 + khalf;
      v2f a[2], b[2];
#pragma unroll
      for (int i = 0; i < 2; ++i)
        a[i] = *(const v2f*)&As[cur][wr * 32 + i * 16 + mrow][kq];
#pragma unroll
      for (int j = 0; j < 2; ++j)
        b[j] = *(const v2f*)&Bs[cur][wc * 32 + j * 16 + mrow][kq];
#pragma unroll
      for (int i = 0; i < 2; ++i)
#pragma unroll
        for (int j = 0; j < 2; ++j)
          acc[i][j] = WMMA_F32(a[i], b[j], acc[i][j]);
    }
    __syncthreads();
  }

  // Epilogue: scatter to Q/K/V head-major buffers.
  const int rsel = (lane >> 4) << 3;  // +8 rows for upper lane half
#pragma unroll
  for (int i = 0; i < 2; ++i)
#pragma unroll
    for (int j = 0; j < 2; ++j) {
      const int m0  = row0 + wr * 32 + i * 16;
      const int fb  = f0 + wc * 32 + j * 16;
      const int sec = fb >> 9;           // 0=q,1=k,2=v
      const int rem = fb & 511;
      const int h   = rem >> 6;
      const int d0  = rem & 63;
      const int bi  = m0 >> 13;
      const int n0  = m0 & (NN - 1);
      float* base = (sec == 0 ? Qb : (sec == 1 ? Kb : Vb)) +
                    ((size_t)(bi * HEADS + h) * NN + n0) * DH;
#pragma unroll
      for (int r = 0; r < 8; ++r)
        base[(size_t)(r + rsel) * DH + d0 + mrow] = acc[i][j][r];
    }
}

// ---------------------------------------------------------------------------
// Kernel 2: dots_partial = k_norm^T @ v_norm over an n-slice (split-K).
// Instance-norm of k/v rows (over d=64) fused in LDS before WMMA.
// 16 waves: wave (wr,wc) owns the 16x16 tile (d_k block, d_v block).
// Double-buffered async loads; 4 async b128 loads per chunk per thread.
// ---------------------------------------------------------------------------
__device__ __forceinline__ void kv_issue(const float* __restrict__ Kb,
                                         const float* __restrict__ Vb,
                                         int bh, int nb,
                                         float (*kS)[68], float (*vS)[68],
                                         int tid) {
  for (int i = tid; i < 64 * 16; i += 512) {
    int r = i >> 4, c4 = (i & 15) << 2;
    size_t off = ((size_t)bh * NN + nb + r) * DH + c4;
    async_ld_b128(&Kb[off], &kS[r][c4]);
    async_ld_b128(&Vb[off], &vS[r][c4]);
  }
}

__global__ __launch_bounds__(512) void k_dots(const float* __restrict__ Kb,
                                              const float* __restrict__ Vb,
                                              float* __restrict__ part) {
  __shared__ float kS[2][64][68];
  __shared__ float vS[2][64][68];
  const int tid  = threadIdx.x;
  const int lane = tid & 31;
  const int w    = tid >> 5;
  const int wr   = w >> 2;   // 0..3
  const int wc   = w & 3;    // 0..3
  const int bh   = blockIdx.x >> 4;
  const int s    = blockIdx.x & (NSPLIT - 1);
  const int khalf = (lane >> 4) << 1;
  const int mcol  = lane & 15;

  v8f acc = {};

  kv_issue(Kb, Vb, bh, s * 512, kS[0], vS[0], tid);

  for (int ch = 0; ch < 8; ++ch) {
    const int cur = ch & 1;
    if (ch + 1 < 8) {
      kv_issue(Kb, Vb, bh, s * 512 + (ch + 1) * 64, kS[cur ^ 1], vS[cur ^ 1], tid);
      wait_async<4>();
    } else {
      wait_async<0>();
    }
    __syncthreads();

    // Fused instance norm: each wave normalizes rows w, w+16, ... (128 rows)
    for (int rr = w; rr < 128; rr += 16) {
      float* rowp = (rr < 64) ? &kS[cur][rr][0] : &vS[cur][rr - 64][0];
      float x0 = rowp[lane * 2];
      float x1 = rowp[lane * 2 + 1];
      float s1 = x0 + x1;
      float s2 = x0 * x0 + x1 * x1;
#pragma unroll
      for (int off = 16; off > 0; off >>= 1) {
        s1 += __shfl_xor(s1, off, 32);
        s2 += __shfl_xor(s2, off, 32);
      }
      const float mu  = s1 * (1.0f / 64.0f);
      const float var = s2 * (1.0f / 64.0f) - mu * mu;  // biased variance
      const float rs  = rsqrtf(var + 1e-5f);
      rowp[lane * 2]     = (x0 - mu) * rs;
      rowp[lane * 2 + 1] = (x1 - mu) * rs;
    }
    __syncthreads();

    // dots[dk][dv] += sum_n kS[n][dk] * vS[n][dv]  (A = k^T, K-dim = n)
    for (int k4 = 0; k4 < 64; k4 += 4) {
      const int kq = k4 + khalf;
      v2f a, b;
      a.x = kS[cur][kq][wr * 16 + mcol];
      a.y = kS[cur][kq + 1][wr * 16 + mcol];
      b.x = vS[cur][kq][wc * 16 + mcol];
      b.y = vS[cur][kq + 1][wc * 16 + mcol];
      acc = WMMA_F32(a, b, acc);
    }
    __syncthreads();
  }

  const int rsel = (lane >> 4) << 3;
  float* dst = part + (size_t)(bh * NSPLIT + s) * (DH * DH);
#pragma unroll
  for (int r = 0; r < 8; ++r)
    dst[(wr * 16 + r + rsel) * DH + wc * 16 + mcol] = acc[r];
}

// ---------------------------------------------------------------------------
// Kernel 3: fold split-K partials deterministically, scale by 1/n.
// ---------------------------------------------------------------------------
__global__ __launch_bounds__(256) void k_dots_reduce(const float* __restrict__ part,
                                                     float* __restrict__ dots) {
  const int t  = blockIdx.x * 256 + threadIdx.x;  // < 32*4096
  const int bh = t >> 12;
  const int ij = t & 4095;
  const float* p = part + (size_t)bh * NSPLIT * (DH * DH) + ij;
  float s = 0.f;
#pragma unroll
  for (int i = 0; i < NSPLIT; ++i) s += p[i * (DH * DH)];
  dots[(size_t)bh * (DH * DH) + ij] = s * (1.0f / (float)NN);
}

// ---------------------------------------------------------------------------
// Kernel 4: att = q @ dots per head, written to [b][n][h*64+e] layout.
// Single-shot async fill (12 b128 loads/thread), then K=64 WMMA loop.
// ---------------------------------------------------------------------------
__global__ __launch_bounds__(256) void k_qd(const float* __restrict__ Qb,
                                            const float* __restrict__ dots,
                                            float* __restrict__ att) {
  __shared__ float qS[128][68];
  __shared__ float dS[64][68];
  const int tid  = threadIdx.x;
  const int lane = tid & 31;
  const int w    = tid >> 5;
  const int wr   = w >> 1;
  const int wc   = w & 1;
  const int bh   = blockIdx.x >> 6;
  const int rt   = blockIdx.x & 63;
  const int n0   = rt * 128;
  const int khalf = (lane >> 4) << 1;
  const int mrow  = lane & 15;

  for (int i = tid; i < 128 * 16; i += 256) {
    int r = i >> 4, c4 = (i & 15) << 2;
    async_ld_b128(&Qb[((size_t)bh * NN + n0 + r) * DH + c4], &qS[r][c4]);
  }
  for (int i = tid; i < 64 * 16; i += 256) {
    int r = i >> 4, c4 = (i & 15) << 2;
    async_ld_b128(&dots[(size_t)bh * (DH * DH) + r * DH + c4], &dS[r][c4]);
  }
  wait_async<0>();
  __syncthreads();

  v8f acc[2][2] = {};
  for (int k4 = 0; k4 < 64; k4 += 4) {
    const int kq = k4 + khalf;
    v2f a[2], b[2];
#pragma unroll
    for (int i = 0; i < 2; ++i)
      a[i] = *(const v2f*)&qS[wr * 32 + i * 16 + mrow][kq];
#pragma unroll
    for (int j = 0; j < 2; ++j) {
      b[j].x = dS[kq][wc * 32 + j * 16 + mrow];
      b[j].y = dS[kq + 1][wc * 32 + j * 16 + mrow];
    }
#pragma unroll
    for (int i = 0; i < 2; ++i)
#pragma unroll
      for (int j = 0; j < 2; ++j)
        acc[i][j] = WMMA_F32(a[i], b[j], acc[i][j]);
  }

  const int rsel = (lane >> 4) << 3;
  const int bi = bh >> 3;
  const int h  = bh & 7;
#pragma unroll
  for (int i = 0; i < 2; ++i)
#pragma unroll
    for (int j = 0; j < 2; ++j) {
      const int nb = n0 + wr * 32 + i * 16;
      const int e0 = wc * 32 + j * 16;
      float* base = att + ((size_t)bi * NN + nb) * INNER + h * DH + e0;
#pragma unroll
      for (int r = 0; r < 8; ++r)
        base[(size_t)(r + rsel) * INNER + mrow] = acc[i][j][r];
    }
}

// ---------------------------------------------------------------------------
// Kernel 5: out = att @ Wout^T + bout (double-buffered async, 8 K-chunks).
// ---------------------------------------------------------------------------
__device__ __forceinline__ void out_issue(const float* __restrict__ att,
                                          const float* __restrict__ Wout,
                                          int row0, int c0, int kk,
                                          float (*As)[68], float (*Bs)[68],
                                          int tid) {
  for (int i = tid; i < 128 * 16; i += 256) {
    int r = i >> 4, c4 = (i & 15) << 2;
    async_ld_b128(&att[(size_t)(row0 + r) * INNER + kk + c4], &As[r][c4]);
  }
  for (int i = tid; i < 64 * 16; i += 256) {
    int r = i >> 4, c4 = (i & 15) << 2;
    async_ld_b128(&Wout[(size_t)(c0 + r) * INNER + kk + c4], &Bs[r][c4]);
  }
}

__global__ __launch_bounds__(256) void k_out(const float* __restrict__ att,
                                             const float* __restrict__ Wout,
                                             const float* __restrict__ bout,
                                             float* __restrict__ out) {
  __shared__ float As[2][128][68];
  __shared__ float Bs[2][64][68];
  const int tid  = threadIdx.x;
  const int lane = tid & 31;
  const int w    = tid >> 5;
  const int wr   = w >> 1;
  const int wc   = w & 1;
  const int rt   = blockIdx.x >> 2;
  const int ct   = blockIdx.x & 3;
  const int row0 = rt * 128;
  const int c0   = ct * 64;
  const int khalf = (lane >> 4) << 1;
  const int mrow  = lane & 15;

  v8f acc[2][2] = {};

  out_issue(att, Wout, row0, c0, 0, As[0], Bs[0], tid);

  for (int c = 0; c < 8; ++c) {
    const int cur = c & 1;
    if (c + 1 < 8) {
      out_issue(att, Wout, row0, c0, (c + 1) * 64, As[cur ^ 1], Bs[cur ^ 1], tid);
      wait_async<12>();
    } else {
      wait_async<0>();
    }
    __syncthreads();

    for (int k4 = 0; k4 < 64; k4 += 4) {
      const int kq = k4 + khalf;
      v2f a[2], b[2];
#pragma unroll
      for (int i = 0; i < 2; ++i)
        a[i] = *(const v2f*)&As[cur][wr * 32 + i * 16 + mrow][kq];
#pragma unroll
      for (int j = 0; j < 2; ++j)
        b[j] = *(const v2f*)&Bs[cur][wc * 32 + j * 16 + mrow][kq];
#pragma unroll
      for (int i = 0; i < 2; ++i)
#pragma unroll
        for (int j = 0; j < 2; ++j)
          acc[i][j] = WMMA_F32(a[i], b[j], acc[i][j]);
    }
    __syncthreads();
  }

  const int rsel = (lane >> 4) << 3;
#pragma unroll
  for (int i = 0; i < 2; ++i)
#pragma unroll
    for (int j = 0; j < 2; ++j) {
      const int m0  = row0 + wr * 32 + i * 16;
      const int col = c0 + wc * 32 + j * 16 + mrow;
      const float bv = bout[col];
#pragma unroll
      for (int r = 0; r < 8; ++r)
        out[(size_t)(m0 + r + rsel) * DIMC + col] = acc[i][j][r] + bv;
    }
}

// ---------------------------------------------------------------------------
extern "C" void kernel_launch(void* const* d_in, const int* in_sizes, int n_in,
                              void* d_out, int out_size, void* d_ws, size_t ws_size,
                              hipStream_t stream) {
  const float* x    = (const float*)d_in[0];
  const float* Wqkv = (const float*)d_in[1];
  const float* Wout = (const float*)d_in[2];
  const float* bout = (const float*)d_in[3];
  float* out = (float*)d_out;

  const size_t SZ = (size_t)BH * NN * DH;  // 16,777,216 floats per q/k/v buffer
  float* Qb   = (float*)d_ws;
  float* Kb   = Qb + SZ;
  float* Vb   = Kb + SZ;
  float* part = Vb + SZ;                               // 32*16*4096 floats (8 MB)
  float* dots = part + (size_t)BH * NSPLIT * DH * DH;  // 131072 floats
  float* att  = Kb;  // K buffer is dead after k_dots_reduce -> reuse for att

  // 1) qkv projection (fp32 WMMA GEMM, async double-buffered), head-major q/k/v
  k_qkv<<<(BN / 128) * 24, 256, 0, stream>>>(x, Wqkv, Qb, Kb, Vb);
  // 2) fused instance-norm + k^T v, split-K over n
  k_dots<<<BH * NSPLIT, 512, 0, stream>>>(Kb, Vb, part);
  // 3) deterministic split-K fold + 1/n scaling
  k_dots_reduce<<<(BH * DH * DH) / 256, 256, 0, stream>>>(part, dots);
  // 4) q @ dots, written in [b][n][inner] layout
  k_qd<<<BH * (NN / 128), 256, 0, stream>>>(Qb, dots, att);
  // 5) output projection + bias
  k_out<<<(BN / 128) * 4, 256, 0, stream>>>(att, Wout, bout, out);
}